// DropEmbedding_45681272160754
// MI455X (gfx1250) — compile-verified
//
#include <hip/hip_runtime.h>

// Problem constants (match reference)
#define SEQ   2048
#define BATCH 8
#define NINP  1024
#define KEEP_E 0.9f     // 1 - DROPOUTE
#define KEEP_I 0.35f    // 1 - DROPOUTI

typedef float v4f __attribute__((ext_vector_type(4)));

// One block (256 threads = 8 wave32) per token (s,b). Each thread produces one
// float4 of the 1024-wide embedding row:
//   - token id + row-dropout uniform are block-uniform -> scalar loads (s_load)
//   - W row gather + u_lock read are 128-bit coalesced vector loads
//   - output is streamed with a non-temporal 128-bit store so the 67MB output
//     doesn't evict the (L2-resident) parts of the 206MB W table.
__global__ __launch_bounds__(256)
void drop_embed_gather_kernel(const int*   __restrict__ X,       // [S*B]
                              const float* __restrict__ W,       // [V, D]
                              const float* __restrict__ u_embed, // [V]
                              const float* __restrict__ u_lock,  // [B, D]
                              float*       __restrict__ out)     // [S*B, D]
{
    const int bid = blockIdx.x;          // flat token index = s*B + b
    const int tid = threadIdx.x;         // 0..255
    const int b   = bid & (BATCH - 1);   // batch index (B == 8, power of two)

    // Block-uniform: token id and its embedding-row dropout draw.
    const int   tok    = X[bid];                         // uniform -> s_load
    const float ue     = u_embed[tok];                   // uniform -> s_load
    const float escale = (ue < KEEP_E) ? (1.0f / KEEP_E) : 0.0f;

    const int d4 = tid << 2;             // starting dim of this thread's float4

    // 128-bit gather of the embedding row slice and the locked-dropout draws.
    const v4f w = *(const v4f*)(W      + (size_t)tok * NINP + d4); // global_load_b128
    const v4f l = *(const v4f*)(u_lock + (size_t)b   * NINP + d4); // global_load_b128 (L2 hit)

    v4f r;
#pragma unroll
    for (int i = 0; i < 4; ++i) {
        const float lscale = (l[i] < KEEP_I) ? (1.0f / KEEP_I) : 0.0f;
        r[i] = w[i] * (escale * lscale);
    }

    // Streaming 128-bit store with non-temporal hint (gfx1250 TH cache control).
    __builtin_nontemporal_store(r, (v4f*)(out + (size_t)bid * NINP + d4));
}

extern "C" void kernel_launch(void* const* d_in, const int* in_sizes, int n_in,
                              void* d_out, int out_size, void* d_ws, size_t ws_size,
                              hipStream_t stream)
{
    // setup_inputs() order: X [S,B] int32, W [V,D] f32, u_embed [V,1] f32, u_lock [1,B,D] f32
    const int*   X       = (const int*)  d_in[0];
    const float* W       = (const float*)d_in[1];
    const float* u_embed = (const float*)d_in[2];
    const float* u_lock  = (const float*)d_in[3];
    float*       out     = (float*)d_out;

    const dim3 grid(SEQ * BATCH);   // 16384 blocks, one per token
    const dim3 block(NINP / 4);     // 256 threads -> one float4 each

    drop_embed_gather_kernel<<<grid, block, 0, stream>>>(X, W, u_embed, u_lock, out);
}